// DistanceEstimator_21990232555679
// MI455X (gfx1250) — compile-verified
//
#include <hip/hip_runtime.h>
#include <hip/hip_bf16.h>

// ---------------------------------------------------------------------------
// Problem constants (from the reference)
// ---------------------------------------------------------------------------
static constexpr int R = 8;       // relations
static constexpr int N = 65536;   // nodes per branch
static constexpr int E = 524288;  // edges per branch
static constexpr int B = 256;     // graphs in batch
static constexpr int C = 128;     // channels (IN == H == 128)

typedef __attribute__((ext_vector_type(2))) float v2f;
typedef __attribute__((ext_vector_type(8))) float v8f;

// ---------------------------------------------------------------------------
// WMMA GEMM: out[row, 0:128] (init/accum) = (A[row,:] * rowscale) @ W (+ bias)
//   A   : [rows, 128] f32 row-major
//   W   : [128, 128]  f32 row-major, staged in LDS k-pair interleaved:
//         sW[(k>>1)*256 + col*2 + (k&1)]  ->  B fragment {k, k+1} is one
//         contiguous 8-byte LDS load (ds_load_b64), bank-conflict-free.
//   cnt : optional per-row divisor (mean aggregation), scale = 1/max(cnt,1)
//   flags bit0: INIT (start from bias / zero) else accumulate from `out`
//   flags bit1: ReLU on store
// Block = 256 threads = 8 waves; each wave computes a 16x128 strip with
// 8 x v_wmma_f32_16x16x4_f32 per K-step, 32 K-steps.
// ---------------------------------------------------------------------------
__global__ __launch_bounds__(256)
void gemm128_wmma_kernel(const float* __restrict__ A,
                         const float* __restrict__ W,
                         const float* __restrict__ cnt,
                         const float* __restrict__ bias,
                         float* __restrict__ out,
                         int flags)
{
    __shared__ float sW[128 * 128];   // 64 KB, exactly fits

    // Cooperative load of W into LDS, k-pair interleaved.
    // Global reads coalesced (consecutive tids -> consecutive cols);
    // LDS writes stride-2 dwords -> banks 2*lane mod 64, conflict-free.
    {
        int tid = threadIdx.x;
#pragma unroll
        for (int i = 0; i < 64; ++i) {
            int idx = i * 256 + tid;          // 0 .. 16383
            int k   = idx >> 7;
            int col = idx & 127;
            sW[(k >> 1) * 256 + col * 2 + (k & 1)] = W[idx];
        }
    }
    __syncthreads();

    const int wave = threadIdx.x >> 5;
    const int lane = threadIdx.x & 31;
    const int l16  = lane & 15;
    const int hi   = lane >> 4;

    const int row_base = blockIdx.x * 128 + wave * 16;
    const int arow     = row_base + l16;          // row this lane feeds to A-frag

    float sc = 1.0f;
    if (cnt) sc = 1.0f / fmaxf(cnt[arow], 1.0f);  // mean = agg / max(cnt,1)

    v8f acc[8];
    if (flags & 1) {
#pragma unroll
        for (int t = 0; t < 8; ++t) {
            float bv = bias ? bias[t * 16 + l16] : 0.0f;  // bias indexed by col
#pragma unroll
            for (int v = 0; v < 8; ++v) acc[t][v] = bv;
        }
    } else {
#pragma unroll
        for (int t = 0; t < 8; ++t)
#pragma unroll
            for (int v = 0; v < 8; ++v)
                acc[t][v] = out[(size_t)(row_base + v + 8 * hi) * C + t * 16 + l16];
    }

    const float* Arow = A + (size_t)arow * C + 2 * hi;
    // LDS base for this lane's B fragments: row-pair (k/2 + hi), l16*32
    const float* sB = sW + (size_t)hi * 256 + l16 * 32;   // + t*32 per tile, + (k/2)*256 per step

#pragma unroll 4
    for (int k = 0; k < 128; k += 4) {
        // A fragment (16x4 f32): lanes 0-15 -> K={k,k+1}, lanes 16-31 -> K={k+2,k+3}
        v2f a = *(const v2f*)(Arow + k);
        a.x *= sc;
        a.y *= sc;
        const float* sBk = sB + (k >> 1) * 256;
#pragma unroll
        for (int t = 0; t < 8; ++t) {
            v2f b = *(const v2f*)(sBk + t * 32);   // {W[k+2hi,col], W[k+2hi+1,col]}
            acc[t] = __builtin_amdgcn_wmma_f32_16x16x4_f32(
                false, a, false, b, (short)0, acc[t], false, false);
        }
    }

    const bool relu = (flags & 2) != 0;
#pragma unroll
    for (int t = 0; t < 8; ++t)
#pragma unroll
        for (int v = 0; v < 8; ++v) {
            float o = acc[t][v];
            if (relu) o = fmaxf(o, 0.0f);
            out[(size_t)(row_base + v + 8 * hi) * C + t * 16 + l16] = o;
        }
}

// ---------------------------------------------------------------------------
// Zero fill
// ---------------------------------------------------------------------------
__global__ void fill_zero_kernel(float* __restrict__ p, int n)
{
    int i = blockIdx.x * blockDim.x + threadIdx.x;
    if (i < n) p[i] = 0.0f;
}

// ---------------------------------------------------------------------------
// Per-relation edge scatter: agg[dst,:] += x[src,:], cnt[dst] += 1
// One wave per edge (grid-stride), 4 float atomics per lane.
// ---------------------------------------------------------------------------
__global__ __launch_bounds__(256)
void scatter_rel_kernel(const float* __restrict__ x, const int* __restrict__ src,
                        const int* __restrict__ dst, const int* __restrict__ et,
                        int ne, int rel,
                        float* __restrict__ agg, float* __restrict__ cnt)
{
    int wid  = (blockIdx.x * blockDim.x + threadIdx.x) >> 5;
    int lane = threadIdx.x & 31;
    int nw   = (gridDim.x * blockDim.x) >> 5;
    for (int e = wid; e < ne; e += nw) {
        if (et[e] != rel) continue;
        int s = src[e], d = dst[e];
        if (lane == 0) atomicAdd(&cnt[d], 1.0f);
        const float* xs = x + (size_t)s * C;
        float*       ad = agg + (size_t)d * C;
#pragma unroll
        for (int j = 0; j < 4; ++j)
            atomicAdd(&ad[lane + 32 * j], xs[lane + 32 * j]);
    }
}

// ---------------------------------------------------------------------------
// Global mean pool: scatter + finalize
// ---------------------------------------------------------------------------
__global__ __launch_bounds__(256)
void pool_scatter_kernel(const float* __restrict__ h, const int* __restrict__ batch,
                         float* __restrict__ pool, float* __restrict__ cntb, int n)
{
    int wid  = (blockIdx.x * blockDim.x + threadIdx.x) >> 5;
    int lane = threadIdx.x & 31;
    if (wid >= n) return;
    int b = batch[wid];
    const float* hr = h + (size_t)wid * C;
    if (lane == 0) atomicAdd(&cntb[b], 1.0f);
#pragma unroll
    for (int j = 0; j < 4; ++j)
        atomicAdd(&pool[b * C + lane + 32 * j], hr[lane + 32 * j]);
}

__global__ void pool_finish_kernel(const float* __restrict__ pool,
                                   const float* __restrict__ cntb,
                                   float* __restrict__ gemb, int coloff)
{
    int i = blockIdx.x * blockDim.x + threadIdx.x;  // 0 .. B*C-1
    int b = i >> 7, c = i & 127;
    gemb[b * (2 * C) + coloff + c] = pool[i] / fmaxf(cntb[b], 1.0f);
}

// ---------------------------------------------------------------------------
// Depth normalization (population std) + build z = [graph_emb, d_norm]  [B,257]
// Single block of 256 threads.
// ---------------------------------------------------------------------------
__global__ __launch_bounds__(256)
void depth_z_kernel(const float* __restrict__ depth, const float* __restrict__ gemb,
                    float* __restrict__ z)
{
    __shared__ float ssum[256];
    __shared__ float ssq[256];
    int b = threadIdx.x;
    float d = depth[b];
    ssum[b] = d;
    ssq[b]  = d * d;
    __syncthreads();
    for (int s = 128; s > 0; s >>= 1) {
        if (b < s) { ssum[b] += ssum[b + s]; ssq[b] += ssq[b + s]; }
        __syncthreads();
    }
    float mean = ssum[0] * (1.0f / 256.0f);
    float var  = ssq[0] * (1.0f / 256.0f) - mean * mean;
    float stdv = sqrtf(fmaxf(var, 0.0f));
    float dn   = (d - mean) / (stdv + 1e-6f);
    for (int j = 0; j < 2 * C; ++j)
        z[b * (2 * C + 1) + j] = gemb[b * (2 * C) + j];
    z[b * (2 * C + 1) + 2 * C] = dn;
}

// ---------------------------------------------------------------------------
// Regression MLP (tiny; bandwidth-trivial)
// ---------------------------------------------------------------------------
__global__ __launch_bounds__(128)
void reg1_kernel(const float* __restrict__ z, const float* __restrict__ W1,
                 const float* __restrict__ b1, float* __restrict__ h1)
{
    int b = blockIdx.x;
    int o = threadIdx.x;
    const float* zb = z + b * (2 * C + 1);
    float acc = b1[o];
    for (int k = 0; k < 2 * C + 1; ++k)
        acc += zb[k] * W1[k * C + o];
    h1[b * C + o] = fmaxf(acc, 0.0f);
}

__global__ __launch_bounds__(256)
void reg2_kernel(const float* __restrict__ h1, const float* __restrict__ W2,
                 const float* __restrict__ b2, float* __restrict__ outp)
{
    int b = threadIdx.x;
    const float* hb = h1 + b * C;
    float acc = b2[0];
    for (int o = 0; o < C; ++o) acc += hb[o] * W2[o];
    outp[b] = acc;
}

// ---------------------------------------------------------------------------
// Host-side sequencing helpers (all launches on `stream`, graph-capture safe)
// ---------------------------------------------------------------------------
static void run_rgcn_layer(const float* X, const float* W, const float* root,
                           const float* bias, const int* src, const int* dst,
                           const int* et, float* outbuf, float* agg, float* cnt,
                           hipStream_t stream)
{
    // out = X @ root + bias   (INIT)
    gemm128_wmma_kernel<<<N / 128, 256, 0, stream>>>(X, root, nullptr, bias, outbuf, 1);
    for (int r = 0; r < R; ++r) {
        fill_zero_kernel<<<(N * C) / 256, 256, 0, stream>>>(agg, N * C);
        fill_zero_kernel<<<N / 256, 256, 0, stream>>>(cnt, N);
        scatter_rel_kernel<<<4096, 256, 0, stream>>>(X, src, dst, et, E, r, agg, cnt);
        // out += (agg / max(cnt,1)) @ W[r] ; ReLU fused on the last relation
        int fl = (r == R - 1) ? 2 : 0;
        gemm128_wmma_kernel<<<N / 128, 256, 0, stream>>>(
            agg, W + (size_t)r * C * C, cnt, nullptr, outbuf, fl);
    }
}

static void run_branch(const float* x, const int* ei, const int* et, const int* batch,
                       const float* W1, const float* root1, const float* b1,
                       const float* W2, const float* root2, const float* b2,
                       float* bufA, float* bufB, float* agg, float* cnt,
                       float* pool, float* cntb, float* gemb, int coloff,
                       hipStream_t stream)
{
    const int* src = ei;        // edge_index[0,:]
    const int* dst = ei + E;    // edge_index[1,:]
    run_rgcn_layer(x,    W1, root1, b1, src, dst, et, bufA, agg, cnt, stream);
    run_rgcn_layer(bufA, W2, root2, b2, src, dst, et, bufB, agg, cnt, stream);
    fill_zero_kernel<<<(B * C) / 256, 256, 0, stream>>>(pool, B * C);
    fill_zero_kernel<<<1, 256, 0, stream>>>(cntb, B);
    pool_scatter_kernel<<<N / 8, 256, 0, stream>>>(bufB, batch, pool, cntb, N);
    pool_finish_kernel<<<(B * C) / 256, 256, 0, stream>>>(pool, cntb, gemb, coloff);
}

extern "C" void kernel_launch(void* const* d_in, const int* in_sizes, int n_in,
                              void* d_out, int out_size, void* d_ws, size_t ws_size,
                              hipStream_t stream)
{
    (void)in_sizes; (void)n_in; (void)out_size; (void)ws_size;

    // ----- inputs (setup_inputs dict order) -----
    const float* state_x     = (const float*)d_in[0];
    const int*   state_ei    = (const int*)  d_in[1];
    const int*   state_et    = (const int*)  d_in[2];
    const int*   state_batch = (const int*)  d_in[3];
    const float* goal_x      = (const float*)d_in[4];
    const int*   goal_ei     = (const int*)  d_in[5];
    const int*   goal_et     = (const int*)  d_in[6];
    const int*   goal_batch  = (const int*)  d_in[7];
    const float* depth       = (const float*)d_in[8];
    const float* state_W1    = (const float*)d_in[9];
    const float* state_root1 = (const float*)d_in[10];
    const float* state_b1    = (const float*)d_in[11];
    const float* state_W2    = (const float*)d_in[12];
    const float* state_root2 = (const float*)d_in[13];
    const float* state_b2    = (const float*)d_in[14];
    const float* goal_W1     = (const float*)d_in[15];
    const float* goal_root1  = (const float*)d_in[16];
    const float* goal_b1     = (const float*)d_in[17];
    const float* goal_W2     = (const float*)d_in[18];
    const float* goal_root2  = (const float*)d_in[19];
    const float* goal_b2     = (const float*)d_in[20];
    const float* reg_W1      = (const float*)d_in[21];
    const float* reg_b1      = (const float*)d_in[22];
    const float* reg_W2      = (const float*)d_in[23];
    const float* reg_b2      = (const float*)d_in[24];

    // ----- workspace layout (floats); peak ~102 MB -----
    float* ws     = (float*)d_ws;
    size_t off    = 0;
    float* bufA   = ws + off; off += (size_t)N * C;       // layer-1 output
    float* bufB   = ws + off; off += (size_t)N * C;       // layer-2 output
    float* agg    = ws + off; off += (size_t)N * C;       // per-relation aggregate
    float* cnt    = ws + off; off += (size_t)N;           // per-relation counts
    float* pool   = ws + off; off += (size_t)B * C;       // pooled sums
    float* cntb   = ws + off; off += (size_t)B;           // nodes per graph
    float* gemb   = ws + off; off += (size_t)B * 2 * C;   // [B, 256] concat embedding
    float* z      = ws + off; off += (size_t)B * (2 * C + 1); // [B, 257]
    float* h1     = ws + off; off += (size_t)B * C;       // reg hidden

    // ----- two encoder branches (sequential, buffers reused) -----
    run_branch(state_x, state_ei, state_et, state_batch,
               state_W1, state_root1, state_b1, state_W2, state_root2, state_b2,
               bufA, bufB, agg, cnt, pool, cntb, gemb, /*coloff=*/0, stream);
    run_branch(goal_x, goal_ei, goal_et, goal_batch,
               goal_W1, goal_root1, goal_b1, goal_W2, goal_root2, goal_b2,
               bufA, bufB, agg, cnt, pool, cntb, gemb, /*coloff=*/C, stream);

    // ----- depth normalization + regression head -----
    depth_z_kernel<<<1, 256, 0, stream>>>(depth, gemb, z);
    reg1_kernel<<<B, 128, 0, stream>>>(z, reg_W1, reg_b1, h1);
    reg2_kernel<<<1, 256, 0, stream>>>(h1, reg_W2, reg_b2, (float*)d_out);
}